// Distance_map_89215060672915
// MI455X (gfx1250) — compile-verified
//
#include <hip/hip_runtime.h>
#include <stdint.h>

// Distance map via exact separable EDT (min-plus with parabola per axis),
// CDNA5 path: Tensor Data Mover (tensor_load_to_lds) stages every line tile
// into LDS asynchronously, tracked with TENSORcnt (s_wait_tensorcnt).
// WMMA is not applicable: min-plus is a (min,+) semiring op, not a matmul.

#define BIGF 1e10f
#define NZ 64
#define NY 128
#define NX 128
#define NTOT (NZ * NY * NX)

typedef uint32_t u32;
typedef __attribute__((ext_vector_type(4))) u32 v4u;
typedef __attribute__((ext_vector_type(8))) u32 v8u;

// Low 32 bits of a flat LDS pointer == LDS byte offset (aperture in high bits).
__device__ __forceinline__ u32 lds_off(const void* p) {
  return (u32)(uintptr_t)p;
}

// TDM 2-D f32 tile load: tile_d0 contiguous elements x tile_d1 rows
// (row stride = row_stride elements) from global -> contiguous LDS.
// tile_d1 == 0 => 1-D load of tile_d0 elements.
// D# per CDNA5 ISA ch.8: group0 = {count/flags, lds_addr, global_addr, type=2},
// group1 = {data_size, tensor dims, tile dims, dim0 stride}. 2-group form
// (VADDR2/3 = NULL) supports tensors up to 2-D.
__device__ __forceinline__ void tdm_load_f32(const float* gsrc, u32 lds_byte,
                                             u32 tile_d0, u32 tile_d1,
                                             u32 row_stride) {
  uint64_t ga = (uint64_t)(uintptr_t)gsrc;
  u32 td1 = tile_d1 ? tile_d1 : 1u;  // tensor_dim1 (avoid 0 for OOB logic)
  v4u g0;
  g0[0] = 1u;                                    // count=1 (valid user D#)
  g0[1] = lds_byte;                              // lds_addr (bytes)
  g0[2] = (u32)ga;                               // global_addr[31:0]
  g0[3] = ((u32)(ga >> 32) & 0x01FFFFFFu)        // global_addr[56:32]
          | (2u << 30);                          // type=2 ("image")
  v8u g1;
  g1[0] = 2u << 16;                              // data_size=2 -> 4 bytes
  g1[1] = (tile_d0 & 0xFFFFu) << 16;             // tensor_dim0[15:0]
  g1[2] = ((tile_d0 >> 16) & 0xFFFFu)            // tensor_dim0[31:16]
          | ((td1 & 0xFFFFu) << 16);             // tensor_dim1[15:0]
  g1[3] = ((td1 >> 16) & 0xFFFFu)                // tensor_dim1[31:16]
          | ((tile_d0 & 0xFFFFu) << 16);         // tile_dim0
  g1[4] = tile_d1 & 0xFFFFu;                     // tile_dim1 (tile_dim2=0)
  g1[5] = row_stride;                            // tensor_dim0_stride[31:0]
  g1[6] = 0u;                                    // stride hi / dim1_stride lo
  g1[7] = 0u;
  asm volatile("tensor_load_to_lds %0, %1" :: "s"(g0), "s"(g1) : "memory");
}

__device__ __forceinline__ void tdm_wait0() {
  asm volatile("s_wait_tensorcnt 0x0" ::: "memory");
}

// ---------------------------------------------------------------- binarize
__global__ __launch_bounds__(256) void k_binarize(const float* __restrict__ x,
                                                  const float* __restrict__ thrp,
                                                  float* __restrict__ fbg,
                                                  float* __restrict__ ffg) {
  int i = blockIdx.x * 256 + threadIdx.x;
  float thr = *thrp;
  float xv = x[i];
  // faithful two-step masking: c[x>=t]=1; c[c<=t]=0  -> b in {0,1}
  float c = (xv >= thr) ? 1.0f : xv;
  float b = (c <= thr) ? 0.0f : c;
  bool fg = (b != 0.0f);
  // EDT(background=1-b): zeros where b==1 ; EDT(foreground=b): zeros where b==0
  fbg[i] = fg ? 0.0f : BIGF;
  ffg[i] = fg ? BIGF : 0.0f;
}

// ------------------------------------------------------- axis X (stride 1)
// 8 waves/block; each wave TDM-loads its own 128-float line into LDS and
// computes d[i] = min_j (i-j)^2 + f[j]; lane owns i = lane + {0,32,64,96}.
__global__ __launch_bounds__(256) void k_pass_x(float* bufA, float* bufB) {
  __shared__ float lds[8 * NX];
  float* buf = (blockIdx.y == 0) ? bufA : bufB;
  const int wave = threadIdx.x >> 5;
  const int lane = threadIdx.x & 31;
  float* line = buf + ((size_t)blockIdx.x * 8 + wave) * NX;
  float* f = &lds[wave * NX];
  tdm_load_f32(line, lds_off(f), NX, 0, NX);
  tdm_wait0();
  float i0 = (float)lane, i1 = i0 + 32.f, i2 = i0 + 64.f, i3 = i0 + 96.f;
  float d0 = BIGF, d1 = BIGF, d2 = BIGF, d3 = BIGF;
#pragma unroll 4
  for (int j = 0; j < NX; ++j) {
    float fj = f[j];  // wave-broadcast LDS read
    float jf = (float)j;
    float t0 = i0 - jf, t1 = i1 - jf, t2 = i2 - jf, t3 = i3 - jf;
    d0 = fminf(d0, fmaf(t0, t0, fj));
    d1 = fminf(d1, fmaf(t1, t1, fj));
    d2 = fminf(d2, fmaf(t2, t2, fj));
    d3 = fminf(d3, fmaf(t3, t3, fj));
  }
  line[lane] = d0;            // coalesced
  line[lane + 32] = d1;
  line[lane + 64] = d2;
  line[lane + 96] = d3;
}

// ----------------------------------------------------- axis Y (stride 128)
// Block: one z-slice, 32-wide x-group. Each wave TDM-loads a private
// 4-column x 128-row strided tile; results go through a 33-padded LDS tile
// (conflict-free column writes) and are stored back coalesced.
__global__ __launch_bounds__(256) void k_pass_y(float* bufA, float* bufB) {
  __shared__ float tin[8][NY * 4];   // per-wave [j][cl]
  __shared__ float tout[NY * 33];    // [y][x] with pad-33 rows
  float* buf = (blockIdx.z == 0) ? bufA : bufB;
  const int wave = threadIdx.x >> 5;
  const int lane = threadIdx.x & 31;
  float* gbase = buf + (size_t)blockIdx.x * (NY * NX) + blockIdx.y * 32;
  tdm_load_f32(gbase + wave * 4, lds_off(&tin[wave][0]), 4, NY, NX);
  tdm_wait0();
  float i0 = (float)lane, i1 = i0 + 32.f, i2 = i0 + 64.f, i3 = i0 + 96.f;
  for (int cl = 0; cl < 4; ++cl) {
    const float* f = &tin[wave][cl];
    float d0 = BIGF, d1 = BIGF, d2 = BIGF, d3 = BIGF;
#pragma unroll 4
    for (int j = 0; j < NY; ++j) {
      float fj = f[j * 4];  // wave-broadcast
      float jf = (float)j;
      float t0 = i0 - jf, t1 = i1 - jf, t2 = i2 - jf, t3 = i3 - jf;
      d0 = fminf(d0, fmaf(t0, t0, fj));
      d1 = fminf(d1, fmaf(t1, t1, fj));
      d2 = fminf(d2, fmaf(t2, t2, fj));
      d3 = fminf(d3, fmaf(t3, t3, fj));
    }
    int col = wave * 4 + cl;
    tout[lane * 33 + col] = d0;          // pad-33: lanes hit distinct banks
    tout[(lane + 32) * 33 + col] = d1;
    tout[(lane + 64) * 33 + col] = d2;
    tout[(lane + 96) * 33 + col] = d3;
  }
  __syncthreads();
  for (int r = wave; r < NY; r += 8)     // coalesced 128B rows
    gbase[(size_t)r * NX + lane] = tout[r * 33 + lane];
}

// --------------------------------------------------- axis Z (stride 16384)
__global__ __launch_bounds__(256) void k_pass_z(float* bufA, float* bufB) {
  __shared__ float tin[8][NZ * 4];
  __shared__ float tout[NZ * 33];
  float* buf = (blockIdx.z == 0) ? bufA : bufB;
  const int wave = threadIdx.x >> 5;
  const int lane = threadIdx.x & 31;
  float* gbase = buf + (size_t)blockIdx.x * NX + blockIdx.y * 32;
  tdm_load_f32(gbase + wave * 4, lds_off(&tin[wave][0]), 4, NZ, NY * NX);
  tdm_wait0();
  float i0 = (float)lane, i1 = i0 + 32.f;
  for (int cl = 0; cl < 4; ++cl) {
    const float* f = &tin[wave][cl];
    float d0 = BIGF, d1 = BIGF;
#pragma unroll 4
    for (int j = 0; j < NZ; ++j) {
      float fj = f[j * 4];
      float jf = (float)j;
      float t0 = i0 - jf, t1 = i1 - jf;
      d0 = fminf(d0, fmaf(t0, t0, fj));
      d1 = fminf(d1, fmaf(t1, t1, fj));
    }
    int col = wave * 4 + cl;
    tout[lane * 33 + col] = d0;
    tout[(lane + 32) * 33 + col] = d1;
  }
  __syncthreads();
  for (int r = wave; r < NZ; r += 8)
    gbase[(size_t)r * (NY * NX) + lane] = tout[r * 33 + lane];
}

// ---------------------------------------------------------------- finalize
__global__ __launch_bounds__(256) void k_final(const float* __restrict__ fbg,
                                               const float* __restrict__ ffg,
                                               float* __restrict__ out) {
  int i = blockIdx.x * 256 + threadIdx.x;
  out[i] = 1.0f - (sqrtf(fbg[i]) + sqrtf(ffg[i]));
}

extern "C" void kernel_launch(void* const* d_in, const int* in_sizes, int n_in,
                              void* d_out, int out_size, void* d_ws, size_t ws_size,
                              hipStream_t stream) {
  const float* x = (const float*)d_in[0];
  const float* thr = (const float*)d_in[1];
  float* fbg = (float*)d_ws;    // 4 MB scratch
  float* ffg = (float*)d_out;   // foreground EDT aliased into output buffer

  k_binarize<<<NTOT / 256, 256, 0, stream>>>(x, thr, fbg, ffg);
  k_pass_x<<<dim3(NTOT / NX / 8, 2, 1), 256, 0, stream>>>(fbg, ffg);
  k_pass_y<<<dim3(NZ, NX / 32, 2), 256, 0, stream>>>(fbg, ffg);
  k_pass_z<<<dim3(NY, NX / 32, 2), 256, 0, stream>>>(fbg, ffg);
  k_final<<<NTOT / 256, 256, 0, stream>>>(fbg, ffg, (float*)d_out);
}